// Featurizer_66778151518549
// MI455X (gfx1250) — compile-verified
//
#include <hip/hip_runtime.h>
#include <hip/hip_bf16.h>
#include <math.h>

// ---------------- problem constants ----------------
#define BB   4
#define NN_  1024
#define KK   30
#define CSPLIT 8   // column-tile slices per 16-row strip (parallelism knob)

typedef __attribute__((ext_vector_type(2))) float v2f;
typedef __attribute__((ext_vector_type(8))) float v8f;

static constexpr long long BN    = (long long)BB * NN_;   // 4096
static constexpr long long BNK   = BN * KK;               // 122880
static constexpr long long OFF_X   = 0;                   // X_out  : BN*12
static constexpr long long OFF_S   = OFF_X + BN * 12;     // S_out  : BN
static constexpr long long OFF_V   = OFF_S + BN;          // _V     : BN*117
static constexpr long long OFF_E   = OFF_V + BN * 117;    // _E     : BNK*272
static constexpr long long OFF_ES  = OFF_E + BNK * 272;   // E_sparse: 2*BNK
static constexpr long long OFF_BID = OFF_ES + 2 * BNK;    // batch_id: BN

// ---------------- small vector helpers ----------------
struct F3 { float x, y, z; };
__device__ __forceinline__ F3 f3sub(F3 a, F3 b) { return F3{a.x - b.x, a.y - b.y, a.z - b.z}; }
__device__ __forceinline__ float f3dot(F3 a, F3 b) { return a.x * b.x + a.y * b.y + a.z * b.z; }
__device__ __forceinline__ F3 f3cross(F3 a, F3 b) {
    return F3{a.y * b.z - a.z * b.y, a.z * b.x - a.x * b.z, a.x * b.y - a.y * b.x};
}
// reference _normalize: zero vector -> zero vector
__device__ __forceinline__ F3 f3nrm(F3 a) {
    float s = f3dot(a, a);
    if (s == 0.0f) return F3{0.f, 0.f, 0.f};
    float inv = 1.0f / sqrtf(s);
    return F3{a.x * inv, a.y * inv, a.z * inv};
}
__device__ __forceinline__ float fsign(float v) { return (v > 0.f) ? 1.f : ((v < 0.f) ? -1.f : 0.f); }

__device__ __forceinline__ F3 atomP(const float* X, int b, int i, int a) {
    const float* p = X + ((size_t)(b * NN_ + i) * 4 + a) * 3;
    return F3{p[0], p[1], p[2]};
}
// backbone chain Xc[p]: residue p/3, atom p%3 (0=N,1=Ca,2=C)
__device__ __forceinline__ F3 chainP(const float* X, int b, int p) { return atomP(X, b, p / 3, p % 3); }
__device__ __forceinline__ F3 Uvec(const float* X, int b, int p) {
    return f3nrm(f3sub(chainP(X, b, p + 1), chainP(X, b, p)));
}
// local frame Q (rows = b1, n0, b1 x n0); zero matrix for last residue (pad row)
__device__ __forceinline__ void frameQ(const float* X, int b, int i, float Q[3][3]) {
    if (i >= NN_ - 1) {
        for (int r = 0; r < 3; r++) for (int c = 0; c < 3; c++) Q[r][c] = 0.f;
        return;
    }
    F3 u0 = Uvec(X, b, 3 * i);
    F3 u1 = Uvec(X, b, 3 * i + 1);
    F3 b1 = f3nrm(f3sub(u0, u1));
    F3 n0 = f3nrm(f3cross(u0, u1));
    F3 r2 = f3cross(b1, n0);
    Q[0][0] = b1.x; Q[0][1] = b1.y; Q[0][2] = b1.z;
    Q[1][0] = n0.x; Q[1][1] = n0.y; Q[1][2] = n0.z;
    Q[2][0] = r2.x; Q[2][1] = r2.y; Q[2][2] = r2.z;
}
__device__ __forceinline__ void rbf16(float d, float* out) {
#pragma unroll
    for (int t = 0; t < 16; t++) {
        float mu = (float)t * (20.0f / 15.0f);     // linspace(0,20,16)
        float e  = (d - mu) * (1.0f / 1.25f);      // sigma = 20/16
        out[t]   = expf(-e * e);
    }
}

// ---------------- kernel 1: Ca pairwise distance via f32 WMMA ----------------
// One wave owns a 16-row strip and sweeps NN_/16/CSPLIT column tiles.
// A operand (row points) + row-norm/mask broadcasts loaded ONCE per wave;
// column loop unrolled x2 -> two v_wmma_f32_16x16x4_f32 in flight.
//   A 16x4 f32 layout: lanes 0-15 hold (K0,K1)=(x,y), lanes 16-31 hold (K2,K3)=(z,0)
//   B 4x16 f32 mirrors it. D^2 = r_i + r_j - 2*G[i][j]; masked pairs get a big
//   sentinel so top-k ordering matches the reference's D_max+1 adjustment.
__global__ void k_dist_wmma(const float* __restrict__ X, const float* __restrict__ mask,
                            float* __restrict__ D) {
    const int lane = threadIdx.x;
    const int half = lane >> 4;
    const int l    = lane & 15;
    const int b    = blockIdx.y;
    const int row  = blockIdx.x * 16 + l;

    F3 pr = atomP(X, b, row, 1);   // Ca of row point
    v2f a;
    a[0] = half ? pr.z : pr.x;
    a[1] = half ? 0.0f : pr.y;
    float rrow = f3dot(pr, pr);
    float mrow = mask[b * NN_ + row];

    // hoist row-side broadcasts out of the column loop (C/D row M = v + 8*half)
    float rr[8], mr[8];
#pragma unroll
    for (int v = 0; v < 8; v++) {
        int M = v + 8 * half;
        rr[v] = __shfl(rrow, M);
        mr[v] = __shfl(mrow, M);
    }

    float* Db = D + ((size_t)b * NN_ + (size_t)blockIdx.x * 16) * NN_;
    const int ct0 = blockIdx.z * (NN_ / 16 / CSPLIT);
    const int ct1 = ct0 + (NN_ / 16 / CSPLIT);

    for (int ct = ct0; ct < ct1; ct += 2) {
#pragma unroll
        for (int u = 0; u < 2; u++) {
            const int colBase = (ct + u) * 16;
            const int col     = colBase + l;
            F3 pc = atomP(X, b, col, 1);
            v2f bm;
            bm[0] = half ? pc.z : pc.x;
            bm[1] = half ? 0.0f : pc.y;
            float rcol = f3dot(pc, pc);
            float mcol = mask[b * NN_ + col];

            v8f acc = {0.f, 0.f, 0.f, 0.f, 0.f, 0.f, 0.f, 0.f};
            acc = __builtin_amdgcn_wmma_f32_16x16x4_f32(false, a, false, bm,
                                                        (short)0, acc, false, false);
#pragma unroll
            for (int v = 0; v < 8; v++) {
                int   M  = v + 8 * half;
                float d2 = rr[v] + rcol - 2.0f * acc[v];
                float d  = sqrtf(fmaxf(d2, 0.0f) + 1e-6f);
                Db[(size_t)M * NN_ + colBase + l] = (mr[v] * mcol != 0.0f) ? d : 3.0e38f;
            }
        }
    }
    __builtin_amdgcn_s_wait_tensorcnt(0);   // CDNA5 split-counter path (no-op here)
}

// ---------------- kernel 2: per-row top-K (wave32, register-resident) --------
// Each lane owns cols {t*32+lane}; used-bitmask keeps v[] in registers.
// Tie-break = lowest index (matches jax.lax.top_k stability).
__global__ void k_topk(const float* __restrict__ D, int* __restrict__ Eidx) {
    const int lane = threadIdx.x;
    const int rowg = blockIdx.x;                 // 0 .. B*N-1
    const float* Dr = D + (size_t)rowg * NN_;
    __builtin_prefetch(Dr + lane, 0, 0);         // global_prefetch_b8 (L2-resident row)
    float v[32];
#pragma unroll
    for (int t = 0; t < 32; t++) v[t] = Dr[t * 32 + lane];
    unsigned used = 0u;

    for (int r = 0; r < KK; r++) {
        float bv = __builtin_huge_valf();
        int   bi = 1 << 30;
#pragma unroll
        for (int t = 0; t < 32; t++) {
            bool ok = ((used >> t) & 1u) == 0u;
            if (ok && v[t] < bv) { bv = v[t]; bi = t * 32 + lane; }
        }
#pragma unroll
        for (int off = 16; off > 0; off >>= 1) {
            float ov = __shfl_xor(bv, off);
            int   oi = __shfl_xor(bi, off);
            if (ov < bv || (ov == bv && oi < bi)) { bv = ov; bi = oi; }
        }
        if ((bi & 31) == lane) used |= 1u << (bi >> 5);
        if (lane == 0) Eidx[(size_t)rowg * KK + r] = bi;
    }
}

// ---------------- kernel 3: batch shifts from mask ----------------
__global__ void k_shift(const float* __restrict__ mask, int* __restrict__ shift) {
    __shared__ float red[256];
    __shared__ float lens[BB];
    for (int b = 0; b < BB; b++) {
        float s = 0.f;
        for (int t = threadIdx.x; t < NN_; t += 256) s += mask[b * NN_ + t];
        red[threadIdx.x] = s;
        __syncthreads();
        for (int o = 128; o > 0; o >>= 1) {
            if (threadIdx.x < o) red[threadIdx.x] += red[threadIdx.x + o];
            __syncthreads();
        }
        if (threadIdx.x == 0) lens[b] = red[0];
        __syncthreads();
    }
    if (threadIdx.x == 0) {
        int acc = 0;
        for (int b = 0; b < BB; b++) { shift[b] = acc; acc += (int)(lens[b] + 0.5f); }
    }
}

// ---------------- kernel 4: node features (_V, X_out, S_out, batch_id) -------
__global__ void k_node(const float* __restrict__ X, const int* __restrict__ S,
                       float* __restrict__ out) {
    int id = blockIdx.x * blockDim.x + threadIdx.x;
    if (id >= (int)BN) return;
    int b = id / NN_, i = id % NN_;
    float* V = out + OFF_V + (size_t)id * 117;

    F3 aN  = atomP(X, b, i, 0), aCa = atomP(X, b, i, 1);
    F3 aC  = atomP(X, b, i, 2), aO  = atomP(X, b, i, 3);

    // V_dist: NODE_PAIRS = (Ca,N)(Ca,C)(Ca,O)(N,C)(N,O)(O,C)
    F3 p0[6] = {aCa, aCa, aCa, aN, aN, aO};
    F3 p1[6] = {aN,  aC,  aO,  aC, aO, aC};
#pragma unroll
    for (int p = 0; p < 6; p++) {
        F3 dd = f3sub(p0[p], p1[p]);
        rbf16(sqrtf(f3dot(dd, dd) + 1e-6f), V + p * 16);
    }

    // V_angles: dihedral + bond-angle cos/sin (arccos folded away analytically)
    float cD[3], sD[3], cA[3], sA[3];
    for (int c = 0; c < 3; c++) {
        int t = 3 * i + c - 1;                      // pad (1,2) over 3n-3 entries
        if (t < 0 || t > 3 * NN_ - 4) { cD[c] = 1.f; sD[c] = 0.f; cA[c] = 1.f; sA[c] = 0.f; continue; }
        F3 u0 = Uvec(X, b, t), u1 = Uvec(X, b, t + 1), u2 = Uvec(X, b, t + 2);
        F3 n0 = f3nrm(f3cross(u0, u1)), n1 = f3nrm(f3cross(u1, u2));
        float cd = fminf(fmaxf(f3dot(n0, n1), -1.0f + 1e-7f), 1.0f - 1e-7f);
        F3 vv = f3nrm(f3cross(n0, n1));
        float sg = fsign(-f3dot(vv, u1));
        cD[c] = cd; sD[c] = sg * sqrtf(fmaxf(1.f - cd * cd, 0.f));
        float ca = fminf(fmaxf(f3dot(u0, u1), -1.0f + 1e-7f), 1.0f - 1e-7f);
        cA[c] = ca; sA[c] = sqrtf(fmaxf(1.f - ca * ca, 0.f));
    }
    float* Va = V + 96;
    Va[0] = cD[0]; Va[1] = cD[1]; Va[2] = cD[2];
    Va[3] = sD[0]; Va[4] = sD[1]; Va[5] = sD[2];
    Va[6] = cA[0]; Va[7] = cA[1]; Va[8] = cA[2];
    Va[9] = sA[0]; Va[10] = sA[1]; Va[11] = sA[2];

    // V_direct: N/C/O relative to Ca, rotated into local frame, normalized
    float Q[3][3];
    frameQ(X, b, i, Q);
    F3 din[3] = {f3sub(aN, aCa), f3sub(aC, aCa), f3sub(aO, aCa)};
    float* Vd = V + 108;
#pragma unroll
    for (int a = 0; a < 3; a++) {
        F3 du = F3{Q[0][0] * din[a].x + Q[0][1] * din[a].y + Q[0][2] * din[a].z,
                   Q[1][0] * din[a].x + Q[1][1] * din[a].y + Q[1][2] * din[a].z,
                   Q[2][0] * din[a].x + Q[2][1] * din[a].y + Q[2][2] * din[a].z};
        F3 nn = f3nrm(du);
        Vd[a * 3 + 0] = nn.x; Vd[a * 3 + 1] = nn.y; Vd[a * 3 + 2] = nn.z;
    }

    // X_out copy, S_out, batch_id (int-valued, exactly representable in f32)
    const float* xp = X + (size_t)id * 12;
    float* xo = out + OFF_X + (size_t)id * 12;
#pragma unroll
    for (int t = 0; t < 12; t++) xo[t] = xp[t];
    out[OFF_S + id]   = (float)S[id];
    out[OFF_BID + id] = (float)b;
}

// ---------------- kernel 5: edge features (_E, E_sparse) ----------------
__global__ void k_edge(const float* __restrict__ X, const int* __restrict__ Eidx,
                       const int* __restrict__ shift, float* __restrict__ out) {
    long long e = (long long)blockIdx.x * blockDim.x + threadIdx.x;
    if (e >= BNK) return;
    int ni = (int)(e / KK);
    int b = ni / NN_, i = ni % NN_;
    int j = Eidx[e];

    float* E = out + OFF_E + (size_t)e * 272;

    float Pi[4][3], Pj[4][3];
#pragma unroll
    for (int a = 0; a < 4; a++)
#pragma unroll
        for (int c = 0; c < 3; c++) {
            Pi[a][c] = X[((size_t)(b * NN_ + i) * 4 + a) * 3 + c];
            Pj[a][c] = X[((size_t)(b * NN_ + j) * 4 + a) * 3 + c];
        }

    // EDGE_PAIRS atom ids: N=0,Ca=1,C=2,O=3
    const int pa[16] = {1, 1, 2, 1, 0, 1, 3, 2, 2, 0, 2, 3, 0, 0, 3, 3};
    const int pc[16] = {1, 2, 1, 0, 1, 3, 1, 2, 0, 2, 3, 2, 0, 3, 0, 3};
#pragma unroll
    for (int p = 0; p < 16; p++) {
        float dx = Pi[pa[p]][0] - Pj[pc[p]][0];
        float dy = Pi[pa[p]][1] - Pj[pc[p]][1];
        float dz = Pi[pa[p]][2] - Pj[pc[p]][2];
        rbf16(sqrtf(dx * dx + dy * dy + dz * dz + 1e-6f), E + p * 16);
    }

    float Qi[3][3], Qj[3][3];
    frameQ(X, b, i, Qi);
    frameQ(X, b, j, Qj);

    // R = Qi^T * Qj  ->  quaternion q (E_angles)
    float R[3][3];
#pragma unroll
    for (int r = 0; r < 3; r++)
#pragma unroll
        for (int c = 0; c < 3; c++)
            R[r][c] = Qi[0][r] * Qj[0][c] + Qi[1][r] * Qj[1][c] + Qi[2][r] * Qj[2][c];
    float Rxx = R[0][0], Ryy = R[1][1], Rzz = R[2][2];
    float m0 = 0.5f * sqrtf(fabsf(1.f + Rxx - Ryy - Rzz));
    float m1 = 0.5f * sqrtf(fabsf(1.f - Rxx + Ryy - Rzz));
    float m2 = 0.5f * sqrtf(fabsf(1.f - Rxx - Ryy + Rzz));
    float q0 = fsign(R[2][1] - R[1][2]) * m0;
    float q1 = fsign(R[0][2] - R[2][0]) * m1;
    float q2 = fsign(R[1][0] - R[0][1]) * m2;
    float qw = 0.5f * sqrtf(fmaxf(1.f + Rxx + Ryy + Rzz, 0.f));
    float qs = q0 * q0 + q1 * q1 + q2 * q2 + qw * qw;
    float qi = (qs == 0.f) ? 0.f : (1.0f / sqrtf(qs));
    E[256] = q0 * qi; E[257] = q1 * qi; E[258] = q2 * qi; E[259] = qw * qi;

    // E_direct: neighbor Ca/N/C/O relative to Ca_i, rotated by Qi, normalized
    const int ord[4] = {1, 0, 2, 3};   // Ca, N, C, O
#pragma unroll
    for (int a = 0; a < 4; a++) {
        F3 d = F3{Pj[ord[a]][0] - Pi[1][0], Pj[ord[a]][1] - Pi[1][1], Pj[ord[a]][2] - Pi[1][2]};
        F3 du = F3{Qi[0][0] * d.x + Qi[0][1] * d.y + Qi[0][2] * d.z,
                   Qi[1][0] * d.x + Qi[1][1] * d.y + Qi[1][2] * d.z,
                   Qi[2][0] * d.x + Qi[2][1] * d.y + Qi[2][2] * d.z};
        F3 nn = f3nrm(du);
        E[260 + a * 3 + 0] = nn.x; E[260 + a * 3 + 1] = nn.y; E[260 + a * 3 + 2] = nn.z;
    }

    // E_sparse: row 0 = dst, row 1 = src
    out[OFF_ES + e]       = (float)(shift[b] + i);
    out[OFF_ES + BNK + e] = (float)(shift[b] + j);
}

// ---------------- launcher ----------------
extern "C" void kernel_launch(void* const* d_in, const int* in_sizes, int n_in,
                              void* d_out, int out_size, void* d_ws, size_t ws_size,
                              hipStream_t stream) {
    const float* X    = (const float*)d_in[0];
    const int*   S    = (const int*)d_in[1];
    const float* mask = (const float*)d_in[2];
    float* out = (float*)d_out;

    // workspace: D matrix (B*N*N f32 = 16.8MB), E_idx (B*N*K i32), shift (B i32)
    float* D     = (float*)d_ws;
    int*   Eidx  = (int*)((char*)d_ws + (size_t)BB * NN_ * NN_ * sizeof(float));
    int*   shift = Eidx + BNK;

    dim3 gd(NN_ / 16, BB, CSPLIT);
    k_dist_wmma<<<gd, 32, 0, stream>>>(X, mask, D);
    k_topk<<<(int)BN, 32, 0, stream>>>(D, Eidx);
    k_shift<<<1, 256, 0, stream>>>(mask, shift);
    k_node<<<(int)((BN + 127) / 128), 128, 0, stream>>>(X, S, out);
    k_edge<<<(int)((BNK + 127) / 128), 128, 0, stream>>>(X, Eidx, shift, out);
}